// SAE_49160195670612
// MI455X (gfx1250) — compile-verified
//
#include <hip/hip_runtime.h>
#include <hip/hip_bf16.h>

typedef __attribute__((ext_vector_type(2))) float v2f;
typedef __attribute__((ext_vector_type(8))) float v8f;

#define D_IN   2048
#define D_SAE  16384
#define BATCH  4096
#define KCAP   64

#define BM 128
#define BN 128
#define KP 16
#define STR 20   // LDS row stride in floats: 80 B -> B128-aligned rows, conflict-free frag reads

// ---- gfx1250 async global->LDS copy (ASYNCcnt path), 16 B per lane ----------
__device__ __forceinline__ void async_copy_b128(unsigned lds_off, const float* g)
{
    asm volatile("global_load_async_to_lds_b128 %0, %1, off"
                 :: "v"(lds_off), "v"(g) : "memory");
}
__device__ __forceinline__ void wait_async0()
{
    asm volatile("s_wait_asynccnt 0x0" ::: "memory");
}
__device__ __forceinline__ unsigned lds_off32(const void* p)
{
    // generic pointers to LDS carry the LDS byte offset in their low 32 bits
    return (unsigned)(unsigned long long)p;
}

// ---------------------------------------------------------------------------
// Kernel A: pre = x @ W_enc^T + b_enc ; codes = relu(pre - tau) ; mask = pre > tau
// 128x128 C-tile / 256 threads (8 wave32, each wave 64x32 via 4x2 v8f accs).
// f32 WMMA 16x16x4. Double-buffered K-panels (KP=16) filled by async-to-LDS DMA:
//   per panel: s_wait_asynccnt 0 -> s_barrier -> issue next panel -> 32 WMMAs.
// ---------------------------------------------------------------------------
__global__ __launch_bounds__(256) void sae_encode(
    const float* __restrict__ x, const float* __restrict__ We,
    const float* __restrict__ be, const float* __restrict__ tau,
    float* __restrict__ codes, float* __restrict__ mask)
{
    __shared__ float As[2][BM * STR];
    __shared__ float Bs[2][BN * STR];

    const int n0  = blockIdx.x * BN;
    const int m0  = blockIdx.y * BM;
    const int tid = threadIdx.x;
    const int lane = tid & 31;
    const int wid  = tid >> 5;
    const int wm = (wid & 1) * 64;   // wave row offset (2 waves in M)
    const int wn = (wid >> 1) * 32;  // wave col offset (4 waves in N)
    const int l16  = lane & 15;
    const int half = lane >> 4;

    v8f acc[4][2];
#pragma unroll
    for (int mt = 0; mt < 4; ++mt)
#pragma unroll
        for (int nt = 0; nt < 2; ++nt)
#pragma unroll
            for (int r = 0; r < 8; ++r) acc[mt][nt][r] = 0.0f;

    // loader mapping: 256 threads x (2 rows) x float4 = 128x16 panel per array
    const int lrow = tid >> 2;        // 0..63
    const int lc4  = (tid & 3) * 4;   // 0,4,8,12

    // prologue: panel 0 into buffer 0
#pragma unroll
    for (int r = 0; r < 2; ++r) {
        const int row = lrow + 64 * r;
        async_copy_b128(lds_off32(&As[0][row * STR + lc4]),
                        x  + (size_t)(m0 + row) * D_IN + lc4);
        async_copy_b128(lds_off32(&Bs[0][row * STR + lc4]),
                        We + (size_t)(n0 + row) * D_IN + lc4);
    }

    int buf = 0;
    for (int k0 = 0; k0 < D_IN; k0 += KP, buf ^= 1) {
        wait_async0();        // own panel copies landed
        __syncthreads();      // everyone's copies landed; everyone done with buf^1

        if (k0 + KP < D_IN) { // issue next panel into the other buffer (overlaps WMMA)
#pragma unroll
            for (int r = 0; r < 2; ++r) {
                const int row = lrow + 64 * r;
                async_copy_b128(lds_off32(&As[buf ^ 1][row * STR + lc4]),
                                x  + (size_t)(m0 + row) * D_IN + k0 + KP + lc4);
                async_copy_b128(lds_off32(&Bs[buf ^ 1][row * STR + lc4]),
                                We + (size_t)(n0 + row) * D_IN + k0 + KP + lc4);
            }
        }

        // ---- 4 k-steps of 4, 8 WMMAs each => 32 v_wmma per panel per wave ----
#pragma unroll
        for (int kq = 0; kq < KP; kq += 4) {
            v2f a[4], b[2];
#pragma unroll
            for (int mt = 0; mt < 4; ++mt)
                a[mt] = *(const v2f*)&As[buf][(wm + mt * 16 + l16) * STR + kq + 2 * half];
#pragma unroll
            for (int nt = 0; nt < 2; ++nt)
                b[nt] = *(const v2f*)&Bs[buf][(wn + nt * 16 + l16) * STR + kq + 2 * half];
#pragma unroll
            for (int mt = 0; mt < 4; ++mt)
#pragma unroll
                for (int nt = 0; nt < 2; ++nt)
                    acc[mt][nt] = __builtin_amdgcn_wmma_f32_16x16x4_f32(
                        false, a[mt], false, b[nt], (short)0, acc[mt][nt], false, false);
        }
    }

    // ---- epilogue: bias, tau threshold, write raw codes + mask ----
#pragma unroll
    for (int nt = 0; nt < 2; ++nt) {
        const int col = n0 + wn + nt * 16 + l16;
        const float bev = be[col];
        const float tv  = tau[col];
#pragma unroll
        for (int mt = 0; mt < 4; ++mt) {
#pragma unroll
            for (int r = 0; r < 8; ++r) {
                const int row = m0 + wm + mt * 16 + r + 8 * half;
                const float pre = acc[mt][nt][r] + bev;
                const float c   = pre - tv;
                const size_t o  = (size_t)row * D_SAE + col;
                codes[o] = c > 0.0f ? c : 0.0f;
                mask[o]  = pre > tv ? 1.0f : 0.0f;
            }
        }
    }
}

// ---------------------------------------------------------------------------
// Kernel T: W_dec [D_IN, D_SAE] -> W_dec^T [D_SAE, D_IN] in workspace (fits L2)
// ---------------------------------------------------------------------------
__global__ __launch_bounds__(256) void wdec_transpose(
    const float* __restrict__ W, float* __restrict__ WT)
{
    __shared__ float tile[32][33];
    const int bs = blockIdx.x * 32;             // s block (D_SAE)
    const int bi = blockIdx.y * 32;             // i block (D_IN)
    const int tx = threadIdx.x & 31;
    const int ty = threadIdx.x >> 5;            // 0..7
#pragma unroll
    for (int r = 0; r < 4; ++r)
        tile[ty + 8 * r][tx] = W[(size_t)(bi + ty + 8 * r) * D_SAE + bs + tx];
    __syncthreads();
#pragma unroll
    for (int r = 0; r < 4; ++r)
        WT[(size_t)(bs + ty + 8 * r) * D_IN + bi + tx] = tile[tx][ty + 8 * r];
}

// ---------------------------------------------------------------------------
// Kernel B: exact kth-largest (k=64) per row via 4-pass byte radix select.
// codes >= 0 => IEEE float ordering == uint32 ordering.
// ---------------------------------------------------------------------------
__global__ __launch_bounds__(256) void sae_select(
    const float* __restrict__ codes, float* __restrict__ thresh)
{
    __shared__ unsigned hist[256];
    __shared__ unsigned s_prefix;
    __shared__ unsigned s_krem;

    const int row = blockIdx.x;
    const int tid = threadIdx.x;
    const float* cr = codes + (size_t)row * D_SAE;

    unsigned v[64];
#pragma unroll 8
    for (int j = 0; j < 64; ++j) v[j] = __float_as_uint(cr[tid + j * 256]);

    if (tid == 0) { s_prefix = 0u; s_krem = KCAP; }

    for (int pass = 0; pass < 4; ++pass) {
        __syncthreads();          // hist free to reuse; s_prefix update visible
        hist[tid] = 0u;
        __syncthreads();
        const int shift = 24 - 8 * pass;
        const unsigned pref = s_prefix;
        for (int j = 0; j < 64; ++j) {
            const unsigned u = v[j];
            const bool match = (pass == 0) || ((u >> (shift + 8)) == pref);
            if (match) atomicAdd(&hist[(u >> shift) & 255u], 1u);
        }
        __syncthreads();
        if (tid == 0) {
            unsigned cum = 0;
            int b = 255;
            for (; b > 0; --b) { cum += hist[b]; if (cum >= s_krem) break; }
            if (cum < s_krem) { cum += hist[0]; b = 0; }
            s_krem  -= (cum - hist[b]);
            s_prefix = (pref << 8) | (unsigned)b;
        }
    }
    __syncthreads();
    if (tid == 0) thresh[row] = __uint_as_float(s_prefix);
}

// ---------------------------------------------------------------------------
// Kernel C: apply top-k mask in place, gather survivors, sparse reconstruction.
// transposed=1: W is W_dec^T [D_SAE, D_IN] (coalesced); else raw W_dec strided.
// ---------------------------------------------------------------------------
#define ACT_CAP 256
__global__ __launch_bounds__(256) void sae_apply_decode(
    float* __restrict__ codes, float* __restrict__ mask,
    const float* __restrict__ thresh, const float* __restrict__ W,
    float* __restrict__ recon, int transposed)
{
    __shared__ float s_val[ACT_CAP];
    __shared__ int   s_idx[ACT_CAP];
    __shared__ int   s_cnt;

    const int row = blockIdx.x;
    const int tid = threadIdx.x;
    const float kth = thresh[row];
    float* cr = codes + (size_t)row * D_SAE;
    float* mr = mask  + (size_t)row * D_SAE;

    if (tid == 0) s_cnt = 0;
    __syncthreads();

    for (int j = 0; j < 64; ++j) {
        const int s = tid + j * 256;
        const float c = cr[s];
        if (c >= kth) {
            if (c > 0.0f) {
                const int p = atomicAdd(&s_cnt, 1);
                if (p < ACT_CAP) { s_val[p] = c; s_idx[p] = s; }
            }
        } else {
            cr[s] = 0.0f;   // zero dropped code
            mr[s] = 0.0f;   // zero dropped mask
        }
    }
    __syncthreads();
    const int n = s_cnt < ACT_CAP ? s_cnt : ACT_CAP;

    float acc[8];
#pragma unroll
    for (int j = 0; j < 8; ++j) acc[j] = 0.0f;

    if (transposed) {
        for (int a = 0; a < n; ++a) {
            const float c = s_val[a];
            const float* wp = W + (size_t)s_idx[a] * D_IN + tid;
#pragma unroll
            for (int j = 0; j < 8; ++j) acc[j] += c * wp[j * 256];
        }
    } else {
        for (int a = 0; a < n; ++a) {
            const float c = s_val[a];
            const int s = s_idx[a];
#pragma unroll
            for (int j = 0; j < 8; ++j)
                acc[j] += c * W[(size_t)(tid + j * 256) * D_SAE + s];
        }
    }
#pragma unroll
    for (int j = 0; j < 8; ++j)
        recon[(size_t)row * D_IN + tid + j * 256] = acc[j];
}

// ---------------------------------------------------------------------------
extern "C" void kernel_launch(void* const* d_in, const int* in_sizes, int n_in,
                              void* d_out, int out_size, void* d_ws, size_t ws_size,
                              hipStream_t stream)
{
    const float* x   = (const float*)d_in[0];
    const float* We  = (const float*)d_in[1];
    const float* be  = (const float*)d_in[2];
    const float* tau = (const float*)d_in[3];
    const float* Wd  = (const float*)d_in[4];
    // d_in[5] = k_cap (fixed 64, baked in as KCAP)

    float* recon = (float*)d_out;                                  // BATCH*D_IN
    float* codes = recon + (size_t)BATCH * D_IN;                   // BATCH*D_SAE
    float* maskp = codes + (size_t)BATCH * D_SAE;                  // BATCH*D_SAE

    float* thresh = (float*)d_ws;                                  // BATCH floats
    float* WT = (float*)((char*)d_ws + 16384);                     // D_SAE*D_IN floats
    const size_t wt_bytes = (size_t)D_SAE * D_IN * sizeof(float);
    const int useT = (ws_size >= 16384 + wt_bytes) ? 1 : 0;

    sae_encode<<<dim3(D_SAE / BN, BATCH / BM), 256, 0, stream>>>(
        x, We, be, tau, codes, maskp);

    if (useT)
        wdec_transpose<<<dim3(D_SAE / 32, D_IN / 32), 256, 0, stream>>>(Wd, WT);

    sae_select<<<BATCH, 256, 0, stream>>>(codes, thresh);

    sae_apply_decode<<<BATCH, 256, 0, stream>>>(
        codes, maskp, thresh, useT ? (const float*)WT : Wd, recon, useT);
}